// SocialRecommender_87866440942240
// MI455X (gfx1250) — compile-verified
//
#include <hip/hip_runtime.h>
#include <hip/hip_bf16.h>

// ---------------------------------------------------------------------------
// SocialRecommender for MI455X (gfx1250, wave32, WMMA).
//
// Pipeline per layer:
//   agg = COO-SpMM scatter (global_atomic_add_f32, L2-resident gathers)
//   c_new = LayerNorm(c_prev + agg @ W^T + b)  via V_WMMA_F32_16X16X4_F32
// Layer outputs are written directly into their d_out slices.
// ---------------------------------------------------------------------------

#define DIM 128
#define NU 100000
#define NI 200000
#define NL 2
#define NE_UI 1600000
#define NE_SOC 800000

typedef __attribute__((ext_vector_type(2))) float v2f;
typedef __attribute__((ext_vector_type(8))) float v8f;

// ---------------------------------------------------------------------------
// SpMM scatter: out[src[e]] += val[e] * x[dst[e]]   (one wave per edge,
// each lane owns 4 contiguous dims -> float4 gather + 4 f32 atomics).
// ---------------------------------------------------------------------------
__global__ __launch_bounds__(256) void spmm_kernel(
    const int* __restrict__ src, const int* __restrict__ dst,
    const float* __restrict__ val, const float* __restrict__ x,
    float* __restrict__ out, int nedges)
{
    int t = blockIdx.x * 256 + threadIdx.x;
    int e = t >> 5;
    if (e >= nedges) return;
    int lane = t & 31;
    int s = src[e];
    int d = dst[e];
    float w = val[e];
    const float4 xd = ((const float4*)(x + (size_t)d * DIM))[lane];
    float* o = out + (size_t)s * DIM + lane * 4;
    atomicAdd(o + 0, w * xd.x);
    atomicAdd(o + 1, w * xd.y);
    atomicAdd(o + 2, w * xd.z);
    atomicAdd(o + 3, w * xd.w);
}

// ---------------------------------------------------------------------------
// Fused  out = LayerNorm(resid + agg @ W^T + bias) * g + beta
//
// W (128x128, row-major [out,in]) is staged once per block into 64KB LDS,
// pre-swizzled into B-fragment order for V_WMMA_F32_16X16X4_F32:
//   element W[o][k]  (B[k][o] of the GEMM):
//     kk = k>>2 (k-step), hi = (k>>1)&1, v = k&1, nt = o>>4, i = o&15
//     word addr = kk*512 + nt*64 + hi*32 + i*2 + v
// so a lane (hi = lane>=16, i = lane&15) fetches its {v0,v1} pair with one
// conflict-free ds_load_b64 per n-tile per k-step.
//
// Each wave computes a 16-row x 128-col tile: 8 v8f accumulators,
// 32 k-steps x 8 WMMAs. Fragment layouts per CDNA5 ISA 7.12.2:
//   A: lane m = lane&15, k = kk*4 + 2*hi + v
//   B: lane n = lane&15, k = kk*4 + 2*hi + v
//   D: m = r + 8*hi, n = nt*16 + (lane&15)
// LN row reduction: 8 nt-partials per (lane, r) then shfl_xor 1/2/4/8
// (stays within the 16-lane half owning that m row).
// ---------------------------------------------------------------------------
__global__ __launch_bounds__(256) void linear_ln_kernel(
    const float* __restrict__ agg,    // [N, 128]
    const float* __restrict__ W,      // [128, 128] (out, in)
    const float* __restrict__ bias,   // [128]
    const float* __restrict__ resid,  // [N, 128]
    const float* __restrict__ g,      // [128]
    const float* __restrict__ beta,   // [128]
    float* __restrict__ out,          // [N, 128]
    int nrows)
{
    __shared__ float lds[DIM * DIM];  // 64 KB

    const int tid = threadIdx.x;
    for (int idx = tid; idx < DIM * DIM; idx += 256) {
        int o = idx >> 7, k = idx & 127;
        int addr = ((k >> 2) * 512) + ((o >> 4) * 64) + (((k >> 1) & 1) * 32)
                 + ((o & 15) * 2) + (k & 1);
        lds[addr] = W[idx];
    }
    __syncthreads();

    const int lane = tid & 31;
    const int wid  = tid >> 5;
    const int hi   = lane >> 4;   // 0: lanes 0-15, 1: lanes 16-31
    const int li   = lane & 15;
    const int ntiles = nrows >> 4;  // all row counts are multiples of 16

    for (int tile = blockIdx.x * 8 + wid; tile < ntiles; tile += gridDim.x * 8) {
        const size_t row0 = (size_t)tile * 16;
        const float* arow = agg + (row0 + li) * DIM + hi * 2;

        v8f acc[8];
#pragma unroll
        for (int nt = 0; nt < 8; ++nt) acc[nt] = (v8f)0.0f;

        for (int kk = 0; kk < 32; ++kk) {
            v2f a = *(const v2f*)(arow + kk * 4);
            const float* bbase = lds + kk * 512 + hi * 32 + li * 2;
#pragma unroll
            for (int nt = 0; nt < 8; ++nt) {
                v2f b = *(const v2f*)(bbase + nt * 64);
                acc[nt] = __builtin_amdgcn_wmma_f32_16x16x4_f32(
                    false, a, false, b, (short)0, acc[nt], false, false);
            }
        }

        // Epilogue: bias + residual, LN statistics per row m = r + 8*hi.
        float s1[8], s2[8];
#pragma unroll
        for (int r = 0; r < 8; ++r) { s1[r] = 0.0f; s2[r] = 0.0f; }

#pragma unroll
        for (int nt = 0; nt < 8; ++nt) {
            int n = nt * 16 + li;
            float bn = bias[n];
#pragma unroll
            for (int r = 0; r < 8; ++r) {
                int m = r + hi * 8;
                float v = acc[nt][r] + bn + resid[(row0 + m) * DIM + n];
                acc[nt][r] = v;
                s1[r] += v;
                s2[r] += v * v;
            }
        }
#pragma unroll
        for (int mask = 1; mask < 16; mask <<= 1) {
#pragma unroll
            for (int r = 0; r < 8; ++r) {
                s1[r] += __shfl_xor(s1[r], mask, 32);
                s2[r] += __shfl_xor(s2[r], mask, 32);
            }
        }
#pragma unroll
        for (int nt = 0; nt < 8; ++nt) {
            int n = nt * 16 + li;
            float gn = g[n], bz = beta[n];
#pragma unroll
            for (int r = 0; r < 8; ++r) {
                int m = r + hi * 8;
                float mean = s1[r] * (1.0f / 128.0f);
                float var  = s2[r] * (1.0f / 128.0f) - mean * mean;
                float inv  = rsqrtf(var + 1e-5f);
                out[(row0 + m) * DIM + n] = (acc[nt][r] - mean) * inv * gn + bz;
            }
        }
    }
}

// ---------------------------------------------------------------------------
extern "C" void kernel_launch(void* const* d_in, const int* in_sizes, int n_in,
                              void* d_out, int out_size, void* d_ws, size_t ws_size,
                              hipStream_t stream)
{
    const float* user_emb = (const float*)d_in[0];
    const float* item_emb = (const float*)d_in[1];
    const int*   ui_src   = (const int*)d_in[2];
    const int*   ui_dst   = (const int*)d_in[3];
    const float* ui_val   = (const float*)d_in[4];
    const int*   soc_src  = (const int*)d_in[5];
    const int*   soc_dst  = (const int*)d_in[6];
    const float* soc_val  = (const float*)d_in[7];
    const float* ui_W     = (const float*)d_in[8];
    const float* ui_b     = (const float*)d_in[9];
    const float* soc_W    = (const float*)d_in[10];
    const float* soc_b    = (const float*)d_in[11];
    const float* ln_g     = (const float*)d_in[12];
    const float* ln_b     = (const float*)d_in[13];

    // d_out layout: ui_layers [3,U,D] | soc_layers [3,U,D] | ci [I,D]
    float* out_ui  = (float*)d_out;
    float* out_soc = out_ui + (size_t)(NL + 1) * NU * DIM;
    float* out_ci  = out_soc + (size_t)(NL + 1) * NU * DIM;
    float* aggbuf  = (float*)d_ws;  // needs I*D*4 = 102.4 MB scratch

    const size_t uBytes = (size_t)NU * DIM * sizeof(float);
    const size_t iBytes = (size_t)NI * DIM * sizeof(float);

    // layer 0 of both output stacks = user_emb
    hipMemcpyAsync(out_ui,  user_emb, uBytes, hipMemcpyDeviceToDevice, stream);
    hipMemcpyAsync(out_soc, user_emb, uBytes, hipMemcpyDeviceToDevice, stream);

    const dim3 blk(256);
    const int spmm_ui_blocks  = (NE_UI * 32 + 255) / 256;   // 200000
    const int spmm_soc_blocks = (NE_SOC * 32 + 255) / 256;  // 100000
    const int lin_blocks = 512;

    for (int i = 0; i < NL; ++i) {
        const float* cu_prev = out_ui + (size_t)i * NU * DIM;
        float*       cu_new  = out_ui + (size_t)(i + 1) * NU * DIM;
        const float* cs_prev = out_soc + (size_t)i * NU * DIM;
        float*       cs_new  = out_soc + (size_t)(i + 1) * NU * DIM;
        const float* ci_prev = (i == 0) ? item_emb : out_ci;

        const float* Wui  = ui_W + (size_t)i * DIM * DIM;
        const float* bui  = ui_b + (size_t)i * DIM;
        const float* Wsoc = soc_W + (size_t)i * DIM * DIM;
        const float* bsoc = soc_b + (size_t)i * DIM;
        const float* g0  = ln_g + (size_t)(2 * i) * DIM;
        const float* be0 = ln_b + (size_t)(2 * i) * DIM;
        const float* g1  = ln_g + (size_t)(2 * i + 1) * DIM;
        const float* be1 = ln_b + (size_t)(2 * i + 1) * DIM;

        // agg_u = spmm(src=ui_src, col=ui_dst) @ ci_prev ; cu update
        hipMemsetAsync(aggbuf, 0, uBytes, stream);
        spmm_kernel<<<spmm_ui_blocks, blk, 0, stream>>>(
            ui_src, ui_dst, ui_val, ci_prev, aggbuf, NE_UI);
        linear_ln_kernel<<<lin_blocks, blk, 0, stream>>>(
            aggbuf, Wui, bui, cu_prev, g0, be0, cu_new, NU);

        // agg_i = spmm(src=ui_dst, col=ui_src) @ cu_prev ; ci update
        // (cu_prev slice is never overwritten, so ordering matches reference)
        hipMemsetAsync(aggbuf, 0, iBytes, stream);
        spmm_kernel<<<spmm_ui_blocks, blk, 0, stream>>>(
            ui_dst, ui_src, ui_val, cu_prev, aggbuf, NE_UI);
        linear_ln_kernel<<<lin_blocks, blk, 0, stream>>>(
            aggbuf, Wui, bui, ci_prev, g1, be1, out_ci, NI);

        // agg_s = spmm(soc) @ cs_prev ; cs update
        hipMemsetAsync(aggbuf, 0, uBytes, stream);
        spmm_kernel<<<spmm_soc_blocks, blk, 0, stream>>>(
            soc_src, soc_dst, soc_val, cs_prev, aggbuf, NE_SOC);
        linear_ln_kernel<<<lin_blocks, blk, 0, stream>>>(
            aggbuf, Wsoc, bsoc, cs_prev, g0, be0, cs_new, NU);
    }
}